// SortSpikesLayer_83150566851383
// MI455X (gfx1250) — compile-verified
//
#include <hip/hip_runtime.h>
#include <cstdint>

// Problem constants (match reference)
#define OH_    60
#define OW_    60
#define NBINS  (OH_ * OW_)      // 3600
#define B_     64
#define S_     4096
#define L_     25
#define CAP_   128

// Chunking for deterministic in-order ranking
#define NCHUNK 16
#define SPC    (S_ / NCHUNK)    // 256 spikes per chunk
#define EPC    (SPC * L_)       // 6400 entries per chunk

// ---------- CDNA5 async global<->LDS helpers (ASYNCcnt-tracked) ----------
__device__ __forceinline__ void async_load_b32_to_lds(unsigned lds_off, unsigned voff,
                                                      unsigned long long sbase) {
    asm volatile("global_load_async_to_lds_b32 %0, %1, %2"
                 :: "v"(lds_off), "v"(voff), "s"(sbase) : "memory");
}
__device__ __forceinline__ void async_store_b128_from_lds(unsigned voff, unsigned lds_off,
                                                          unsigned long long sbase) {
    asm volatile("global_store_async_from_lds_b128 %0, %1, %2"
                 :: "v"(voff), "v"(lds_off), "s"(sbase) : "memory");
}
__device__ __forceinline__ void wait_async0() {
    asm volatile("s_wait_asynccnt 0" ::: "memory");
}
__device__ __forceinline__ unsigned lds_off_of(const void* p) {
    return (unsigned)(uintptr_t)p;   // low 32 bits of LDS pointer = byte offset
}

// ---------------- Kernel 1: per-(batch,chunk) bin histogram ----------------
__global__ __launch_bounds__(256) void k_count(const int* __restrict__ spikes,
                                               const int* __restrict__ indices,
                                               int* __restrict__ cnt) {
    __shared__ int hist[NBINS];
    __shared__ int sid[SPC];
    const int b     = blockIdx.x / NCHUNK;
    const int chunk = blockIdx.x % NCHUNK;
    const int tid   = threadIdx.x;

    for (int i = tid; i < NBINS; i += 256) hist[i] = 0;

    // async-copy this chunk's 256 spike ids into LDS (no VGPR round trip)
    {
        unsigned long long srow =
            (unsigned long long)(uintptr_t)(spikes + (size_t)b * S_ + (size_t)chunk * SPC);
        async_load_b32_to_lds(lds_off_of(&sid[tid]), (unsigned)(tid * 4), srow);
        wait_async0();
    }
    __syncthreads();

    for (int e = tid; e < EPC; e += 256) {
        const int ls = e / L_;
        const int l  = e % L_;
        const int id = sid[ls];
        const int g  = indices[id * L_ + l];
        if (g >= 0) atomicAdd(&hist[g % NBINS], 1);
    }
    __syncthreads();

    // async-store the LDS histogram to workspace, 16B at a time (NBINS % 4 == 0)
    unsigned long long wrow =
        (unsigned long long)(uintptr_t)(cnt + (size_t)(b * NCHUNK + chunk) * NBINS);
    for (int i = tid * 4; i < NBINS; i += 256 * 4) {
        async_store_b128_from_lds((unsigned)(i * 4), lds_off_of(&hist[i]), wrow);
    }
    wait_async0();
}

// -------- Kernel 2: exclusive prefix over chunks per (batch, bin) --------
__global__ __launch_bounds__(256) void k_scan(int* __restrict__ cnt) {
    const int t = blockIdx.x * blockDim.x + threadIdx.x;
    if (t >= B_ * NBINS) return;
    const int b   = t / NBINS;
    const int bin = t % NBINS;
    int* p = cnt + (size_t)b * NCHUNK * NBINS + bin;   // adjacent threads -> adjacent addrs
    int run = 0;
    for (int c = 0; c < NCHUNK; ++c) {
        const int v = p[(size_t)c * NBINS];
        p[(size_t)c * NBINS] = run;
        run += v;
    }
}

// ------- Kernel 3: fill output with -1 via async stores from LDS -------
__global__ __launch_bounds__(256) void k_fill(int* __restrict__ out, long long n4) {
    __shared__ int pat[256 * 4];
    const int tid = threadIdx.x;
    pat[tid * 4 + 0] = -1; pat[tid * 4 + 1] = -1;
    pat[tid * 4 + 2] = -1; pat[tid * 4 + 3] = -1;
    __syncthreads();
    const unsigned lds = lds_off_of(&pat[tid * 4]);
    const unsigned long long base = (unsigned long long)(uintptr_t)out;
    const long long stride = (long long)gridDim.x * blockDim.x;
    for (long long i = (long long)blockIdx.x * blockDim.x + tid; i < n4; i += stride) {
        async_store_b128_from_lds((unsigned)(i * 16), lds, base);  // < 2^31, fits I32 voff
    }
    wait_async0();
}

// --- Kernel 4: order-exact scatter; one wave per (batch, chunk), one spike/step ---
//
// Within a spike, the 25 bins (h-kh)*60 + (w-kw) are guaranteed distinct, so all
// 25 lanes can ds_add_rtn concurrently and the returned value is the exact rank.
// Steps execute in spike order and per-wave LDS ops are in-order, so cross-spike
// ordering is exact; cross-chunk ordering comes from the scanned base counters.
__global__ __launch_bounds__(32) void k_scatter(const int* __restrict__ spikes,
                                                const int* __restrict__ indices,
                                                const int* __restrict__ cnt,
                                                int* __restrict__ out) {
    __shared__ int basec[NBINS];   // running rank counter per bin (starts at chunk base)
    __shared__ int sid[SPC];
    const int b     = blockIdx.x / NCHUNK;
    const int chunk = blockIdx.x % NCHUNK;
    const int lane  = threadIdx.x;              // blockDim.x == 32 == one wave

    // async-load base counters + chunk spike ids into LDS
    unsigned long long crow =
        (unsigned long long)(uintptr_t)(cnt + (size_t)(b * NCHUNK + chunk) * NBINS);
    for (int i = lane; i < NBINS; i += 32)
        async_load_b32_to_lds(lds_off_of(&basec[i]), (unsigned)(i * 4), crow);
    unsigned long long srow =
        (unsigned long long)(uintptr_t)(spikes + (size_t)b * S_ + (size_t)chunk * SPC);
    for (int i = lane; i < SPC; i += 32)
        async_load_b32_to_lds(lds_off_of(&sid[i]), (unsigned)(i * 4), srow);
    wait_async0();   // single wave: no block barrier needed

    int* __restrict__ obase = out + (size_t)b * CAP_ * NBINS;
    const bool mylane = (lane < L_);

    // software pipeline, depth 2: overlap the indices-row gather with the
    // previous spike's LDS atomic + scattered store
    int g0 = mylane ? indices[(size_t)sid[0] * L_ + lane] : -1;
    int g1 = mylane ? indices[(size_t)sid[1] * L_ + lane] : -1;

    for (int s = 0; s < SPC; ++s) {
        const int g = g0;
        g0 = g1;
        g1 = (s + 2 < SPC && mylane) ? indices[(size_t)sid[s + 2] * L_ + lane] : -1;

        if (g >= 0) {
            const int bin  = g % NBINS;
            const int ckk  = g / NBINS;
            const int slot = atomicAdd(&basec[bin], 1);   // ds_add_rtn_u32, exact rank
            if (slot < CAP_)
                obase[(size_t)slot * NBINS + bin] = ckk;
        }
    }
}

extern "C" void kernel_launch(void* const* d_in, const int* in_sizes, int n_in,
                              void* d_out, int out_size, void* d_ws, size_t ws_size,
                              hipStream_t stream) {
    const int* spikes  = (const int*)d_in[0];   // (B, S, 1, 1) int32
    const int* indices = (const int*)d_in[1];   // (N_IN, 25) int32
    int* out = (int*)d_out;                     // (B, CAP, OH, OW) int32
    int* cnt = (int*)d_ws;                      // B*NCHUNK*NBINS ints (~14.7 MB)

    k_count  <<<B_ * NCHUNK, 256, 0, stream>>>(spikes, indices, cnt);
    k_scan   <<<(B_ * NBINS + 255) / 256, 256, 0, stream>>>(cnt);
    k_fill   <<<1024, 256, 0, stream>>>(out, (long long)out_size / 4);
    k_scatter<<<B_ * NCHUNK, 32, 0, stream>>>(spikes, indices, cnt, out);
}